// CFVAE_59047210385791
// MI455X (gfx1250) — compile-verified
//
#include <hip/hip_runtime.h>

typedef __attribute__((ext_vector_type(2))) float    v2f;
typedef __attribute__((ext_vector_type(8))) float    v8f;
typedef __attribute__((ext_vector_type(4))) unsigned v4u;
typedef __attribute__((ext_vector_type(8))) unsigned v8u;

__device__ __forceinline__ unsigned rfl_u32(unsigned x) {
  return (unsigned)__builtin_amdgcn_readfirstlane((int)x);
}
__device__ __forceinline__ float rfl_f32(float x) {
  return __int_as_float(__builtin_amdgcn_readfirstlane(__float_as_int(x)));
}

// ---------------------------------------------------------------------------
// Tensor Data Mover: issue a 1-row 2D descriptor load of `tile_elems` f32
// elements from contiguous global memory into LDS. OOB elements (beyond
// tensor_elems) read as zero per ISA §8.2. Caller must be a single wave,
// then s_wait_tensorcnt(0) + __syncthreads() before LDS is consumed.
// D# layout per CDNA5 ISA §8.3 (group0) / §8.4 (group1), 2-group form (2D).
// All inputs are block-uniform; readfirstlane legalizes the VALU->SGPR copies
// required by the explicit SGPR-tuple asm constraints (VIMAGE needs SGPRs).
// ---------------------------------------------------------------------------
__device__ __forceinline__ void tdm_load_f32(void* lds_dst, const void* gsrc,
                                             unsigned tile_elems, unsigned tensor_elems) {
  const unsigned lds_off = rfl_u32((unsigned)(uintptr_t)lds_dst);  // low 32b = group-seg offset
  const unsigned long long ga = (unsigned long long)(uintptr_t)gsrc;
  const unsigned galo = rfl_u32((unsigned)ga);
  const unsigned gahi = rfl_u32((unsigned)(ga >> 32));
  const unsigned tile = rfl_u32(tile_elems);
  const unsigned tens = rfl_u32(tensor_elems);
  v4u g0;
  g0[0] = 1u;                                                   // count=1, user D#
  g0[1] = lds_off;                                              // lds_addr
  g0[2] = galo;                                                 // global_addr[31:0]
  g0[3] = (gahi & 0x01FFFFFFu) | (2u << 30);                    // global_addr[56:32], type=2
  v8u g1;
  g1[0] = 0x00020000u;                                          // data_size=2 (4B), no mcast
  g1[1] = (tens & 0xFFFFu) << 16;                               // tensor_dim0[15:0]
  g1[2] = (tens >> 16) | (1u << 16);                            // tensor_dim0[31:16], tensor_dim1=1
  g1[3] = (tile & 0xFFFFu) << 16;                               // tile_dim0 (<=65535)
  g1[4] = 1u;                                                   // tile_dim1=1, tile_dim2=0
  g1[5] = tens;                                                 // tensor_dim0_stride[31:0]
  g1[6] = 0u;                                                   // stride hi / dim1_stride lo
  g1[7] = 0u;
  asm volatile("tensor_load_to_lds %0, %1"
               : : "{s[36:39]}"(g0), "{s[40:47]}"(g1) : "memory");
}

// Native no-return f32 add atomic, device scope (acks at L2 = coherence point
// for the next kernel's plain loads). Avoids any CAS-loop expansion risk.
__device__ __forceinline__ void atomic_add_f32_dev(float* p, float v) {
  asm volatile("global_atomic_add_f32 %0, %1, off scope:SCOPE_DEV"
               : : "v"(p), "v"(v) : "memory");
}

// ---------------------------------------------------------------------------
// Zero a float buffer (atomic-accumulation targets must start at 0 each call)
// ---------------------------------------------------------------------------
__global__ void zero_kernel(float* __restrict__ p, size_t n) {
  size_t i = (size_t)blockIdx.x * blockDim.x + threadIdx.x;
  size_t stride = (size_t)gridDim.x * blockDim.x;
  for (; i < n; i += stride) p[i] = 0.0f;
}

// ---------------------------------------------------------------------------
// One-block prep: Cinv = inv(I - A^T) via Gauss-Jordan in LDS,
// then M = A^T @ Cinv  (folds the DAG solve + mask into one 32x32 matrix)
// ---------------------------------------------------------------------------
__global__ void prep_kernel(const float* __restrict__ A, float* __restrict__ Mout) {
  __shared__ float B[32][64];             // augmented [ (I-A^T) | I ]
  const int tid = threadIdx.x;            // 256 threads
  const int r  = tid & 31;                // row
  const int jg = tid >> 5;                // col group: cols jg, jg+8, ..., jg+56
#pragma unroll
  for (int t = 0; t < 8; ++t) {
    int j = jg + t * 8;
    float v;
    if (j < 32) v = (r == j ? 1.0f : 0.0f) - A[j * 32 + r];   // (I - A^T)[r][j]
    else        v = ((j - 32) == r) ? 1.0f : 0.0f;
    B[r][j] = v;
  }
  __syncthreads();
  for (int p = 0; p < 32; ++p) {
    const float piv = B[p][p];
    const float f   = B[r][p];            // snapshot before any writes this round
    __syncthreads();
    if (r == p) {
      const float inv = 1.0f / piv;
#pragma unroll
      for (int t = 0; t < 8; ++t) { int j = jg + t * 8; B[p][j] *= inv; }
    }
    __syncthreads();
    if (r != p) {
#pragma unroll
      for (int t = 0; t < 8; ++t) { int j = jg + t * 8; B[r][j] -= f * B[p][j]; }
    }
    __syncthreads();
  }
  // M[c][k] = sum_j A^T[c][j] * Cinv[j][k] = sum_j A[j][c] * B[j][32+k]
  for (int q = tid; q < 1024; q += 256) {
    const int c = q >> 5, k = q & 31;
    float s = 0.0f;
    for (int j = 0; j < 32; ++j) s += A[j * 32 + c] * B[j][32 + k];
    Mout[q] = s;
  }
}

// ---------------------------------------------------------------------------
// Dense GEMM via V_WMMA_F32_16X16X4_F32:  Y[N,128] = act(Xin[N,K]) @ W[K,128]
// Block = 256 threads = 8 waves; block owns a 16-node row stripe, each wave a
// 16-wide column tile. W (16-64KB) staged into LDS by a single TDM descriptor.
// ---------------------------------------------------------------------------
template <int K, bool RELU>
__global__ void gemm_wmma_kernel(const float* __restrict__ Xin,
                                 const float* __restrict__ W,
                                 float* __restrict__ Y, int Nrows) {
  __shared__ float wlds[K * 128];
  const int tid = threadIdx.x;
  if (tid < 32) {                          // wave 0: one TDM op stages all of W
    tdm_load_f32(wlds, W, K * 128, K * 128);
    __builtin_amdgcn_s_wait_tensorcnt(0);
  }
  __syncthreads();

  const int wave = tid >> 5;
  const int lane = tid & 31;
  const int mrow = lane & 15;             // M (A) / N (B,C) index within tile
  const int hi   = lane >> 4;             // lane-half selector
  const int n0   = blockIdx.x * 16;
  const int c0   = wave * 16;

  int arowidx = n0 + mrow;
  if (arowidx >= Nrows) arowidx = Nrows - 1;       // safe dup read, store guarded
  const float* arow = Xin + (size_t)arowidx * K;

  v8f acc = {};
#pragma unroll 4
  for (int kk = 0; kk < K / 4; ++kk) {
    const int kb = kk * 4 + 2 * hi;                // K = v + 2*(lane/16)  (ISA §7.12.2)
    float a0 = arow[kb + 0];
    float a1 = arow[kb + 1];
    if (RELU) { a0 = fmaxf(a0, 0.0f); a1 = fmaxf(a1, 0.0f); }
    v2f a; a[0] = a0; a[1] = a1;
    v2f b; b[0] = wlds[(kb + 0) * 128 + c0 + mrow];
           b[1] = wlds[(kb + 1) * 128 + c0 + mrow];
    acc = __builtin_amdgcn_wmma_f32_16x16x4_f32(
        /*neg_a=*/false, a, /*neg_b=*/false, b,
        /*c_mod=*/(short)0, acc, /*reuse_a=*/false, /*reuse_b=*/false);
  }
  // C/D layout: VGPR i holds rows i (lanes 0-15) and i+8 (lanes 16-31)
  float* ycol = Y + (size_t)n0 * 128 + c0 + mrow;
  if (n0 + 16 <= Nrows) {                  // block-uniform fast path: no EXEC churn
#pragma unroll
    for (int i = 0; i < 8; ++i) ycol[(size_t)(i + 8 * hi) * 128] = acc[i];
  } else {
#pragma unroll
    for (int i = 0; i < 8; ++i) {
      const int row = n0 + i + 8 * hi;
      if (row < Nrows) Y[(size_t)row * 128 + c0 + mrow] = acc[i];
    }
  }
}

// ---------------------------------------------------------------------------
// Edge-parallel spmm: one wave32 per edge. Edge metadata is wave-uniform ->
// readfirstlane forces scalar (SMEM) loads; lanes carry only the coalesced
// 512B row gather + 4 native f32 atomics each (L2-resident on 192MB L2).
// out[dst] += val * x[src]   (out must be pre-zeroed)
// ---------------------------------------------------------------------------
__global__ void spmm_kernel(const int* __restrict__ src, const int* __restrict__ dst,
                            const float* __restrict__ val, const float* __restrict__ x,
                            float* __restrict__ out, int E) {
  const int tid  = blockIdx.x * 256 + threadIdx.x;
  const int e    = __builtin_amdgcn_readfirstlane(tid >> 5);   // wave-uniform edge id
  const int lane = tid & 31;
  if (e >= E) return;
  const int   s = __builtin_amdgcn_readfirstlane(src[e]);
  const int   d = __builtin_amdgcn_readfirstlane(dst[e]);
  const float v = rfl_f32(val[e]);
  const float4 xv = *reinterpret_cast<const float4*>(x + (size_t)s * 128 + lane * 4);
  float* o = out + (size_t)d * 128 + lane * 4;
  atomic_add_f32_dev(o + 0, v * xv.x);
  atomic_add_f32_dev(o + 1, v * xv.y);
  atomic_add_f32_dev(o + 2, v * xv.z);
  atomic_add_f32_dev(o + 3, v * xv.w);
}

// ---------------------------------------------------------------------------
// Fused tail: block = 8 nodes x 32 concepts. e_m slab + M staged by TDM.
//   masked = M @ e_m ; z = g_z(masked) + sqrt(1e-3)*noise
//   rec_x = z.reshape(128) @ Wrec + brec ; pred_label = g_label(label @ A)
// ---------------------------------------------------------------------------
__global__ void tail_kernel(const float* __restrict__ e_m, const float* __restrict__ label,
                            const float* __restrict__ noise, const float* __restrict__ A,
                            const float* __restrict__ Wz1, const float* __restrict__ bz1,
                            const float* __restrict__ Wz2, const float* __restrict__ bz2,
                            const float* __restrict__ Wl1, const float* __restrict__ bl1,
                            const float* __restrict__ Wl2, const float* __restrict__ bl2,
                            const float* __restrict__ Wrec, const float* __restrict__ brec,
                            const float* __restrict__ Mmat, float* __restrict__ out, int N) {
  __shared__ float e_lds[8 * 128];
  __shared__ float z_lds[8 * 128];
  __shared__ float lab_lds[8 * 32];
  __shared__ float M_lds[32 * 32];

  const int tid = threadIdx.x;                 // 256
  const size_t nbase = (size_t)blockIdx.x * 8;

  if (tid < 32) {                              // wave 0: TDM-stage e_m slab + M
    const size_t total = (size_t)N * 128;
    const size_t off   = nbase * 128;
    const unsigned avail = (unsigned)((off < total) ? (total - off < 1024 ? total - off : 1024) : 0);
    tdm_load_f32(e_lds, e_m + off, 1024, avail);   // OOB tail reads as zero
    tdm_load_f32(M_lds, Mmat, 1024, 1024);
    __builtin_amdgcn_s_wait_tensorcnt(0);
  }
  {
    size_t gi = nbase * 32 + tid;
    lab_lds[tid] = (gi < (size_t)N * 32) ? label[gi] : 0.0f;
  }
  __syncthreads();

  const int nl = tid >> 5;                     // node within block
  const int c  = tid & 31;                     // concept
  const size_t node = nbase + nl;
  const bool live = node < (size_t)N;

  // masked[c][:] = sum_k M[c][k] * e_m[n][k][:]
  float md[4] = {0.f, 0.f, 0.f, 0.f};
  const float* er = e_lds + nl * 128;
  for (int k = 0; k < 32; ++k) {
    const float m = M_lds[c * 32 + k];
    md[0] += m * er[k * 4 + 0];
    md[1] += m * er[k * 4 + 1];
    md[2] += m * er[k * 4 + 2];
    md[3] += m * er[k * 4 + 3];
  }
  // g_z: Linear(4,32) -> ELU -> Linear(32,4), weights for concept c
  float zd[4] = {0.f, 0.f, 0.f, 0.f};
  for (int g = 0; g < 32; ++g) {
    float t = bz1[c * 32 + g];
#pragma unroll
    for (int d = 0; d < 4; ++d) t += md[d] * Wz1[(c * 4 + d) * 32 + g];
    t = t > 0.0f ? t : (__expf(t) - 1.0f);     // ELU
#pragma unroll
    for (int d = 0; d < 4; ++d) zd[d] += t * Wz2[(c * 32 + g) * 4 + d];
  }
  const float sqlam = 0.031622776601683794f;   // sqrt(1e-3)
#pragma unroll
  for (int d = 0; d < 4; ++d) {
    const float nz = live ? noise[(node * 32 + c) * 4 + d] : 0.0f;
    z_lds[nl * 128 + c * 4 + d] = zd[d] + bz2[c * 4 + d] + sqlam * nz;
  }

  // pred_label: ml = (label @ A)[n][c]; per-concept Linear(1,32)-ELU-Linear(32,1)
  float ml = 0.0f;
  for (int k = 0; k < 32; ++k) ml += A[k * 32 + c] * lab_lds[nl * 32 + k];
  float pred = bl2[c];
  for (int g = 0; g < 32; ++g) {
    float t = ml * Wl1[c * 32 + g] + bl1[c * 32 + g];
    t = t > 0.0f ? t : (__expf(t) - 1.0f);
    pred += t * Wl2[c * 32 + g];
  }
  if (live) out[(size_t)N * 32 + node * 32 + c] = pred;
  __syncthreads();

  // rec_x[n][i] = sum_h z[n][h] * Wrec[h][i] + brec[i]   (thread: n=nl, i=c)
  float rec = brec[c];
  const float* zr = z_lds + nl * 128;
  for (int h = 0; h < 128; ++h) rec += zr[h] * Wrec[h * 32 + c];
  if (live) out[node * 32 + c] = rec;
}

// ---------------------------------------------------------------------------
extern "C" void kernel_launch(void* const* d_in, const int* in_sizes, int n_in,
                              void* d_out, int out_size, void* d_ws, size_t ws_size,
                              hipStream_t stream) {
  const float* X     = (const float*)d_in[0];
  const float* label = (const float*)d_in[1];
  const float* eval  = (const float*)d_in[2];
  const float* noise = (const float*)d_in[3];
  const float* Wbase = (const float*)d_in[4];
  const float* Wmean = (const float*)d_in[5];
  // d_in[6] = W_logstd: its spmm result is discarded in the reference -> skipped
  const float* A    = (const float*)d_in[7];
  const float* Wz1  = (const float*)d_in[8];
  const float* bz1  = (const float*)d_in[9];
  const float* Wz2  = (const float*)d_in[10];
  const float* bz2  = (const float*)d_in[11];
  const float* Wl1  = (const float*)d_in[12];
  const float* bl1  = (const float*)d_in[13];
  const float* Wl2  = (const float*)d_in[14];
  const float* bl2  = (const float*)d_in[15];
  const float* Wrec = (const float*)d_in[16];
  const float* brec = (const float*)d_in[17];
  const int* esrc   = (const int*)d_in[18];
  const int* edst   = (const int*)d_in[19];
  float* out = (float*)d_out;

  const int N = in_sizes[0] / 32;
  const int E = in_sizes[2];

  float* bufA = (float*)d_ws;                  // [N,128]  XW, then HM
  float* bufB = bufA + (size_t)N * 128;        // [N,128]  hidden-acc, then e_m-acc
  float* Mmat = bufB + (size_t)N * 128;        // [32,32]  A^T (I - A^T)^-1

  // 1) fold DAG solve + mask into M
  prep_kernel<<<1, 256, 0, stream>>>(A, Mmat);
  // 2) hidden = relu(spmm(X @ W_base))
  zero_kernel<<<2048, 256, 0, stream>>>(bufB, (size_t)N * 128);
  gemm_wmma_kernel<32, false><<<(N + 15) / 16, 256, 0, stream>>>(X, Wbase, bufA, N);
  spmm_kernel<<<(E * 32 + 255) / 256, 256, 0, stream>>>(esrc, edst, eval, bufA, bufB, E);
  // 3) e_m = spmm(relu(hidden) @ W_mean)
  gemm_wmma_kernel<128, true><<<(N + 15) / 16, 256, 0, stream>>>(bufB, Wmean, bufA, N);
  zero_kernel<<<2048, 256, 0, stream>>>(bufB, (size_t)N * 128);
  spmm_kernel<<<(E * 32 + 255) / 256, 256, 0, stream>>>(esrc, edst, eval, bufA, bufB, E);
  // 4) fused per-node tail -> rec_x, pred_label
  tail_kernel<<<(N + 7) / 8, 256, 0, stream>>>(bufB, label, noise, A, Wz1, bz1, Wz2, bz2,
                                               Wl1, bl1, Wl2, bl2, Wrec, brec, Mmat, out, N);
}